// HungarianMatcher_Patch_61624190763777
// MI455X (gfx1250) — compile-verified
//
#include <hip/hip_runtime.h>
#include <math.h>

// Problem constants (from reference): bs=2, Q=128, T=64, P=100, 6 classes.
#define Q_DIM   128
#define T_DIM   64
#define P_DIM   100
#define N_TYPES 6
#define NTILE   7          // ceil(100/16)
#define T2S_W   0.2f
#define EPS_V   1e-6f
#define BIGV    1e30f      // padding sentinel (stays finite under fp32 adds)

typedef float v2f __attribute__((ext_vector_type(2)));
typedef float v8f __attribute__((ext_vector_type(8)));

// One wave32 per (b, q, t) pair. 8 waves per 256-thread block.
__global__ __launch_bounds__(256)
void hungarian_matcher_cost_kernel(
    const float* __restrict__ pred_pts,    // [bs,Q,P,3]
    const float* __restrict__ pred_logits, // [bs,Q,2]
    const float* __restrict__ pred_type,   // [bs,Q,6]
    const float* __restrict__ tgt_pts,     // [bs,T,P,3]
    const float* __restrict__ area_w,      // [bs,T]
    const int*   __restrict__ tgt_labels,  // [bs,T]
    float*       __restrict__ out,         // [bs,Q,T]
    int bs)
{
    const int lane = threadIdx.x & 31;
    const int sub  = lane & 15;   // N (or M) index within a 16-wide tile
    const int half = lane >> 4;   // 0: K=0,1 / rows 0-7   1: K=2,3 / rows 8-15

    const int wave  = blockIdx.x * (blockDim.x >> 5) + (threadIdx.x >> 5);
    const int total = bs * Q_DIM * T_DIM;
    if (wave >= total) return;    // wave-uniform: EXEC stays all-ones per wave

    int b   = wave / (Q_DIM * T_DIM);
    int rem = wave - b * Q_DIM * T_DIM;
    int q   = rem / T_DIM;
    int t   = rem - q * T_DIM;

    // ---- preload pred (B-matrix) tiles: B col_j = [-2x,-2y,-2z, 1], plus pn_j ----
    const float* pbase = pred_pts + (size_t)(b * Q_DIM + q) * P_DIM * 3;
    v2f   Bv[NTILE];
    float pnv[NTILE];
#pragma unroll
    for (int tj = 0; tj < NTILE; ++tj) {
        int j = tj * 16 + sub;
        float x = 0.f, y = 0.f, z = 0.f, pn = BIGV;
        if (j < P_DIM) {
            x = pbase[j * 3 + 0];
            y = pbase[j * 3 + 1];
            z = pbase[j * 3 + 2];
            pn = x * x + y * y + z * z;
        }
        v2f bb;
        if (half == 0) { bb.x = -2.f * x; bb.y = -2.f * y; }   // K=0,1 rows
        else           { bb.x = -2.f * z; bb.y = 1.f;      }   // K=2,3 rows
        Bv[tj]  = bb;
        pnv[tj] = pn;
    }

    float colAcc[NTILE];           // running min over all target rows, per pred col
#pragma unroll
    for (int tj = 0; tj < NTILE; ++tj) colAcc[tj] = 4.f * BIGV;

    float s2t_acc = 0.f;
    const float* tbase = tgt_pts + (size_t)(b * T_DIM + t) * P_DIM * 3;

    for (int ti = 0; ti < NTILE; ++ti) {
        // ---- load tgt (A-matrix) tile: A row_i = [x,y,z, tn_i] ----
        int i = ti * 16 + sub;
        float x = 0.f, y = 0.f, z = 0.f, tn = BIGV;
        if (i < P_DIM) {
            x = tbase[i * 3 + 0];
            y = tbase[i * 3 + 1];
            z = tbase[i * 3 + 2];
            tn = x * x + y * y + z * z;
        }
        v2f Av;
        if (half == 0) { Av.x = x; Av.y = y;  }  // K=0,1
        else           { Av.x = z; Av.y = tn; }  // K=2,3

        float rowAcc[8];
#pragma unroll
        for (int r = 0; r < 8; ++r) rowAcc[r] = 4.f * BIGV;

#pragma unroll
        for (int tj = 0; tj < NTILE; ++tj) {
            // C operand pre-loaded with pn_j (per-lane column constant):
            // D[i][j] = tn_i - 2*(t_i . p_j) + pn_j  == full squared distance
            float pn = pnv[tj];
            v8f c = {pn, pn, pn, pn, pn, pn, pn, pn};
            c = __builtin_amdgcn_wmma_f32_16x16x4_f32(
                    false, Av, false, Bv[tj], (short)0, c, false, false);
            float m = c[0];
#pragma unroll
            for (int r = 0; r < 8; ++r) {
                float d = c[r];                      // clamp deferred (monotone)
                rowAcc[r] = fminf(rowAcc[r], d);
                m = fminf(m, d);
            }
            colAcc[tj] = fminf(colAcc[tj], m);
        }

        // row mins: reduce across the 16 lanes of each half (same row set)
#pragma unroll
        for (int r = 0; r < 8; ++r) {
            float v = rowAcc[r];
            v = fminf(v, __shfl_xor(v, 1, 32));
            v = fminf(v, __shfl_xor(v, 2, 32));
            v = fminf(v, __shfl_xor(v, 4, 32));
            v = fminf(v, __shfl_xor(v, 8, 32));
            v = fmaxf(v, 0.f);                       // deferred clamp, once per row
            int row = ti * 16 + half * 8 + r;
            s2t_acc += (row < P_DIM) ? v : 0.f;      // select, no EXEC change
        }
    }
    // halves hold sums over disjoint row subsets -> combine
    s2t_acc += __shfl_xor(s2t_acc, 16, 32);
    float s2t = s2t_acc * (1.f / (float)P_DIM);

    // col mins: lane n and n+16 hold the same column (different row halves)
    float t2s_acc = 0.f;
#pragma unroll
    for (int tj = 0; tj < NTILE; ++tj) {
        float cm = fminf(colAcc[tj], __shfl_xor(colAcc[tj], 16, 32));
        cm = fmaxf(cm, 0.f);                         // deferred clamp, once per col
        int j = tj * 16 + sub;
        t2s_acc += (j < P_DIM) ? cm : 0.f;
    }
    t2s_acc += __shfl_xor(t2s_acc, 1, 32);
    t2s_acc += __shfl_xor(t2s_acc, 2, 32);
    t2s_acc += __shfl_xor(t2s_acc, 4, 32);
    t2s_acc += __shfl_xor(t2s_acc, 8, 32);
    float t2s = t2s_acc * (1.f / (float)P_DIM);

    float w    = area_w[b * T_DIM + t];
    float geom = (s2t + T2S_W * t2s) * (1.f / (1.f + T2S_W)) * w;

    // ---- classification cost (tiny; computed redundantly by all lanes) ----
    const float* lg = pred_logits + (size_t)(b * Q_DIM + q) * 2;
    float l0 = lg[0], l1 = lg[1];
    float lm = fmaxf(l0, l1);
    float e0 = __expf(l0 - lm), e1 = __expf(l1 - lm);
    float vp0 = e0 / (e0 + e1);

    const float* ty = pred_type + (size_t)(b * Q_DIM + q) * N_TYPES;
    int lab = tgt_labels[b * T_DIM + t];
    float tv[N_TYPES];
    float tmax = -3.0e38f;
#pragma unroll
    for (int c2 = 0; c2 < N_TYPES; ++c2) { tv[c2] = ty[c2]; tmax = fmaxf(tmax, tv[c2]); }
    float esum = 0.f, esel = 0.f;
#pragma unroll
    for (int c2 = 0; c2 < N_TYPES; ++c2) {
        float e = __expf(tv[c2] - tmax);
        esum += e;
        esel = (c2 == lab) ? e : esel;
    }
    float tp = esel / esum;
    float cost_class = -__logf(tp + EPS_V) - __logf(vp0 + EPS_V);

    float C = geom + cost_class;
    if (lane == 0) out[(size_t)(b * Q_DIM + q) * T_DIM + t] = C;
}

extern "C" void kernel_launch(void* const* d_in, const int* in_sizes, int n_in,
                              void* d_out, int out_size, void* d_ws, size_t ws_size,
                              hipStream_t stream) {
    const float* pred_pts    = (const float*)d_in[0]; // [bs,Q,P,3]
    const float* pred_logits = (const float*)d_in[1]; // [bs,Q,2]
    const float* pred_type   = (const float*)d_in[2]; // [bs,Q,6]
    const float* tgt_pts     = (const float*)d_in[3]; // [bs,T,P,3]
    const float* area_w      = (const float*)d_in[4]; // [bs,T]
    const int*   tgt_labels  = (const int*)d_in[5];   // [bs,T]
    float* out = (float*)d_out;

    int bs = in_sizes[1] / (Q_DIM * 2);               // from pred_patch_logits

    int totalWaves    = bs * Q_DIM * T_DIM;           // one wave per (b,q,t)
    int wavesPerBlock = 8;                            // 256 threads
    int blocks = (totalWaves + wavesPerBlock - 1) / wavesPerBlock;

    hungarian_matcher_cost_kernel<<<blocks, 256, 0, stream>>>(
        pred_pts, pred_logits, pred_type, tgt_pts, area_w, tgt_labels, out, bs);
}